// EV2VoxelGrid_25658134626635
// MI455X (gfx1250) — compile-verified
//
#include <hip/hip_runtime.h>
#include <hip/hip_bf16.h>
#include <stdint.h>

// EV2VoxelGrid for MI455X (gfx1250, wave32).
// Memory-bound scatter-add: zero-fill 590MB output, then 2 no-return
// global_atomic_add_f32 per valid event. Event stream staged through LDS
// with the CDNA5 async copy engine (ASYNCcnt) in a double buffer.

namespace {
constexpr int kH = 720;
constexpr int kW = 1280;
constexpr int kT = 20;
constexpr int kN = 500000;          // events per batch
constexpr int kHW = kH * kW;        // 921,600
constexpr int kTHW = kT * kHW;      // 18,432,000 (per-batch output stride)
constexpr int kThreads = 256;       // 8 wave32 per block
constexpr int kMaxBlocks = 3125;    // 15625 tiles / 3125 = 5 tiles per block
}

__global__ __launch_bounds__(256) void ev2vox_zero(float4* __restrict__ out, int n4) {
  int i = blockIdx.x * blockDim.x + threadIdx.x;
  const int stride = gridDim.x * blockDim.x;
  const float4 z = make_float4(0.f, 0.f, 0.f, 0.f);
  for (; i < n4; i += stride) out[i] = z;   // global_store_b128, RT (keep lines in L2)
}

// CDNA5 async copy: one 16B event per lane, global -> LDS, tracked by ASYNCcnt.
// NT hint: the 64MB event stream is read once; don't evict output lines from L2.
__device__ __forceinline__ void ev_async_load_b128(unsigned ldsByteAddr, const float4* gsrc) {
  asm volatile("global_load_async_to_lds_b128 %0, %1, off th:TH_LOAD_NT"
               :: "v"(ldsByteAddr), "v"((unsigned long long)(uintptr_t)gsrc)
               : "memory");
}

// No-return f32 atomic add at device scope (TH[0]=0 -> non-returning, STOREcnt).
__device__ __forceinline__ void ev_atomic_add_f32(float* addr, float v) {
  asm volatile("global_atomic_add_f32 %0, %1, off scope:SCOPE_DEV"
               :: "v"((unsigned long long)(uintptr_t)addr), "v"(v)
               : "memory");
}

__global__ __launch_bounds__(256) void ev2vox_scatter(
    const float4* __restrict__ events,   // (B, N, 4) f32 == B*N float4
    const int* __restrict__ counts,      // (B,) int32 (JAX x64 disabled)
    float* __restrict__ out,             // (B, T, H, W) f32
    int numEvents, int numTiles) {
  __shared__ float4 buf[2][kThreads];
  const int tid = threadIdx.x;
  // Generic LDS pointer low 32 bits == workgroup-relative LDS byte offset.
  const unsigned lds0 = (unsigned)(uintptr_t)(&buf[0][tid]);
  const unsigned lds1 = (unsigned)(uintptr_t)(&buf[1][tid]);

  int tile = (int)blockIdx.x;
  {
    // Prefetch first tile into buffer 0 (clamp keeps EXEC all-ones at the tail).
    const int e = min(tile * kThreads + tid, numEvents - 1);
    ev_async_load_b128(lds0, events + e);
  }
  unsigned sel = 0u;
  while (tile < numTiles) {
    // Current tile's async copy is the only outstanding one -> wait to zero.
    asm volatile("s_wait_asynccnt 0x0" ::: "memory");
    const float4 ev = buf[sel][tid];                 // ds_load_b128
    // Ensure the LDS read has retired before the async engine may overwrite it.
    asm volatile("s_wait_dscnt 0x0" ::: "memory");

    const int nextTile = tile + (int)gridDim.x;      // uniform across the block
    if (nextTile < numTiles) {
      const int e = min(nextTile * kThreads + tid, numEvents - 1);
      ev_async_load_b128(sel ? lds0 : lds1, events + e);  // overlaps the atomics below
    }

    const int e = tile * kThreads + tid;
    const int b = e / kN;
    const int idxInB = e - b * kN;
    if (e < numEvents && idxInB < counts[b]) {
      int x = (int)ev.x; x = max(0, min(x, kW - 1));
      int y = (int)ev.y; y = max(0, min(y, kH - 1));
      const float pol = ev.w * 2.0f - 1.0f;          // {0,1} -> {-1,+1}
      const float tb  = ev.z * (float)(kT - 1);
      const float kf  = floorf(tb);
      const int   k0  = (int)kf;
      const float w1  = tb - kf;
      const float w0  = 1.0f - w1;
      const int k0c = max(0, min(k0,     kT - 1));
      const int k1c = max(0, min(k0 + 1, kT - 1));
      // output layout (B, T, H, W): ((b*T + k)*H + y)*W + x
      const int base = b * kTHW + y * kW + x;        // fits int32 (max 147,455,999)
      ev_atomic_add_f32(out + base + k0c * kHW, pol * w0);
      ev_atomic_add_f32(out + base + k1c * kHW, pol * w1);
    }
    sel ^= 1u;
    tile = nextTile;
  }
}

extern "C" void kernel_launch(void* const* d_in, const int* in_sizes, int n_in,
                              void* d_out, int out_size, void* d_ws, size_t ws_size,
                              hipStream_t stream) {
  const float4* events = (const float4*)d_in[0];   // (B, N, 4) float32
  const int*    counts = (const int*)d_in[1];      // (B,) int32
  float*        out    = (float*)d_out;            // (B, T, H, W) float32

  // 1) Zero the 590MB output grid with b128 stores (out_size % 4 == 0 here).
  const int n4 = out_size / 4;
  ev2vox_zero<<<8192, kThreads, 0, stream>>>((float4*)out, n4);

  // 2) Scatter-add all events.
  const int numEvents = in_sizes[0] / 4;                          // B*N = 4,000,000
  const int numTiles  = (numEvents + kThreads - 1) / kThreads;    // 15,625
  const int blocks    = numTiles < kMaxBlocks ? numTiles : kMaxBlocks;
  ev2vox_scatter<<<blocks, kThreads, 0, stream>>>(events, counts, out,
                                                  numEvents, numTiles);
}